// StochVolSimulator_37769942401738
// MI455X (gfx1250) — compile-verified
//
#include <hip/hip_runtime.h>
#include <hip/hip_bf16.h>
#include <stdint.h>

typedef __attribute__((ext_vector_type(2))) float v2f;
typedef __attribute__((ext_vector_type(8))) float v8f;
typedef __attribute__((ext_vector_type(4))) unsigned int u32x4;
typedef __attribute__((ext_vector_type(4))) int i32x4;
typedef __attribute__((ext_vector_type(8))) int i32x8;

constexpr int kTile    = 32768;            // elements per block tile
constexpr int kThreads = 256;              // 8 waves (wave32)
constexpr int kChunk   = 2048;             // elements scanned per block iteration
constexpr int kWaves   = kThreads / 32;
constexpr int kIters   = kTile / kChunk;

#if __has_builtin(__builtin_amdgcn_tensor_load_to_lds) && \
    __has_builtin(__builtin_amdgcn_s_wait_tensorcnt)
#define SV_USE_TDM 1
#else
#define SV_USE_TDM 0
#endif

__device__ __forceinline__ uint32_t sv_hash(uint32_t x) {
  x ^= x >> 16; x *= 0x7feb352dU;
  x ^= x >> 15; x *= 0x846ca68bU;
  x ^= x >> 16;
  return x;
}

__device__ __forceinline__ float sv_wave_reduce(float v) {
#pragma unroll
  for (int o = 16; o > 0; o >>= 1) v += __shfl_down(v, o, 32);
  return v;
}

// ---------- Kernel A: per-tile sums of eps = eps_raw / tau ----------
__global__ __launch_bounds__(kThreads)
void sv_tile_reduce(const float* __restrict__ eps_raw,
                    const float* __restrict__ log_theta,
                    float* __restrict__ part, int n) {
  const float invTau = 1.0f / (__expf(log_theta[1]) + 1.0f);
  const int t = threadIdx.x;
  const int base = blockIdx.x * kTile;
  float s = 0.0f;
  const float4* e4 = reinterpret_cast<const float4*>(eps_raw);
#pragma unroll 4
  for (int k = 0; k < kTile / (kThreads * 4); ++k) {
    int i = base + (k * kThreads + t) * 4;
    if (i + 3 < n) {
      float4 v = e4[i >> 2];
      s += (v.x + v.y) + (v.z + v.w);
    } else {
      for (int u = 0; u < 4; ++u)
        if (i + u < n) s += eps_raw[i + u];
    }
  }
  __shared__ float red[kWaves];
  s = sv_wave_reduce(s);
  if ((t & 31) == 0) red[t >> 5] = s;
  __syncthreads();
  if (t == 0) {
    float tot = 0.0f;
    for (int w = 0; w < kWaves; ++w) tot += red[w];
    part[blockIdx.x] = tot * invTau;
  }
}

// ---------- Kernel B: exclusive scan of tile sums (single block) ----------
__global__ __launch_bounds__(1024)
void sv_scan_partials(const float* __restrict__ part,
                      float* __restrict__ offs, int nb) {
  __shared__ float a[1024];
  __shared__ float carrySh;
  const int t = threadIdx.x;
  if (t == 0) carrySh = 0.0f;
  __syncthreads();
  for (int c0 = 0; c0 < nb; c0 += 1024) {
    const float carry = carrySh;
    const int i = c0 + t;
    const float v = (i < nb) ? part[i] : 0.0f;
    a[t] = v;
    __syncthreads();
    for (int s = 1; s < 1024; s <<= 1) {
      float u = (t >= s) ? a[t - s] : 0.0f;
      __syncthreads();
      a[t] += u;
      __syncthreads();
    }
    if (i < nb) offs[i] = carry + a[t] - v;   // exclusive tile prefix
    __syncthreads();
    if (t == 0) carrySh = carry + a[1023];
    __syncthreads();
  }
}

#if SV_USE_TDM
// LDS byte offset of a __shared__ object (AS3 pointers are 32-bit offsets).
__device__ __forceinline__ uint32_t sv_lds_offset(const void* p) {
  return (uint32_t)(unsigned long long)(__attribute__((address_space(3))) const void*)p;
}

// Issue one TDM DMA: 2048 f32 from global -> LDS (2048x1 tile, 4B elements).
// D# bitfields per CDNA5 ISA ch.8: group0 {count|lds_addr|global_addr|type=2},
// group1 {data_size=4B, tensor_dim0=tile_dim0=2048, tensor_dim1=tile_dim1=1}.
__device__ __forceinline__ void sv_tdm_load_chunk(uint32_t lds_byte_addr,
                                                  const float* gsrc) {
  const unsigned long long ga = (unsigned long long)(uintptr_t)gsrc;
  u32x4 g0;
  g0.x = 1u;                                      // count=1, user descriptor
  g0.y = lds_byte_addr;                           // lds_addr (bytes)
  g0.z = (unsigned)(ga & 0xFFFFFFFFu);            // global_addr[31:0]
  g0.w = (unsigned)((ga >> 32) & 0x01FFFFFFu)     // global_addr[56:32]
       | (2u << 30);                              // type = 2 ("image")
  i32x8 g1;
  g1[0] = (2 << 16);                              // wg_mask=0, data_size=4B
  g1[1] = (int)((kChunk & 0xFFFF) << 16);         // tensor_dim0[15:0] @ bits 63:48
  g1[2] = (int)((kChunk >> 16) | (1 << 16));      // tensor_dim0[31:16] | tensor_dim1[15:0]=1
  g1[3] = (int)(kChunk << 16);                    // tensor_dim1[31:16]=0 | tile_dim0=2048
  g1[4] = 1;                                      // tile_dim1=1, tile_dim2=0
  g1[5] = kChunk;                                 // tensor_dim0_stride[31:0]
  g1[6] = 0;                                      // stride0[47:32] | stride1[15:0]
  g1[7] = 0;                                      // stride1[47:16]
  i32x4 z4 = {0, 0, 0, 0};
#if __clang_major__ >= 23
  i32x8 z8 = {};
  __builtin_amdgcn_tensor_load_to_lds(g0, g1, z4, z4, z8, 0);
#else
  __builtin_amdgcn_tensor_load_to_lds(g0, g1, z4, z4, 0);
#endif
}
#endif  // SV_USE_TDM

// ---------- shared chunk worker: WMMA scan + StudentT emit ----------
// bufc holds 2048 RAW eps values; the 1/tau scale is folded into the WMMA
// A-matrix (triangular entries = invTau), so D = (L*invTau) * X = scan(eps/tau).
template <bool FULL>
__device__ __forceinline__ float sv_chunk_compute(
    const float* __restrict__ bufc, float* __restrict__ sbuf,
    float* __restrict__ wsum, const v2f (&A)[4], float blockOffset,
    int chunkBase, int n, float nu, float whC, float whS,
    float* __restrict__ out) {
  const int t    = threadIdx.x;
  const int lane = t & 31;
  const int wv   = t >> 5;
  const int lj   = lane & 15;   // column (N) index
  const int hi   = lane >> 4;   // half-wave select

  // per-wave 256-elem inclusive scan: D = (invTau*L)(16x16) * X via four
  // chained V_WMMA_F32_16X16X4_F32 (K-chunks of 4, accumulate in D).
  const float* x = bufc + wv * 256;          // X[k][j] = x[j*16 + k]
  v8f acc = {};
#pragma unroll
  for (int c = 0; c < 4; ++c) {
    const int k0 = 4 * c + 2 * hi;
    v2f b;
    b.x = x[lj * 16 + k0];                   // X[k0][lj]
    b.y = x[lj * 16 + k0 + 1];               // X[k0+1][lj]
    acc = __builtin_amdgcn_wmma_f32_16x16x4_f32(
        false, A[c], false, b, (short)0, acc, false, false);
  }
  // acc[r] = inclusive value of chunk element lj*16 + (r + 8*hi) (scaled).

  // cross-column offsets: scan the 16 column totals with lane shuffles.
  float colTot = __shfl(acc[7], 16 + lj, 32);          // D[15][lj]
  float sc = colTot;
#pragma unroll
  for (int s = 1; s < 16; s <<= 1) {
    float u = __shfl(sc, (lane - s) & 31, 32);
    if (lj >= s) sc += u;
  }
  const float colExcl = sc - colTot;
  const float waveTot = __shfl(sc, (lane & 16) + 15, 32);

  if (lane == 0) wsum[wv] = waveTot;
  __syncthreads();
  float wexcl = 0.0f, chunkTot = 0.0f;
#pragma unroll
  for (int w = 0; w < kWaves; ++w) {
    float v = wsum[w];
    if (w < wv) wexcl += v;
    chunkTot += v;
  }

  const float add = blockOffset + wexcl + colExcl;
  float* sb = sbuf + wv * 256;
#pragma unroll
  for (int r = 0; r < 8; ++r)
    sb[lj * 16 + r + 8 * hi] = acc[r] + add;
  __syncthreads();

  // emit: counter-based RNG -> StudentT(nu); x = t * exp(s) * mask factor.
#pragma unroll
  for (int q = 0; q < kChunk / kThreads; ++q) {
    const int e = chunkBase + q * kThreads + t;
    if (FULL || e < n) {
      float s_t = sbuf[q * kThreads + t];
      uint32_t h1 = sv_hash(2u * (uint32_t)e + 0x9e3779b9u);
      uint32_t h2 = sv_hash(2u * (uint32_t)e + 0x6a09e667u);
      float u1 = ((float)h1 + 1.0f) * 2.3283064e-10f;   // (0,1]
      float u2 = (float)h2 * 2.3283064e-10f;            // [0,1)
      float r  = __fsqrt_rn(-2.0f * __logf(u1));
      float sn, cn;
      __sincosf(6.2831853f * u2, &sn, &cn);
      float zn = r * cn;                                 // N(0,1)
      float zg = r * sn;                                 // N(0,1) indep.
      float w  = 1.0f - whC + zg * whS;                  // Wilson–Hilferty
      float chi2 = fmaxf(nu * w * w * w, 1e-20f);        // ~ChiSq(nu)
      float tstd = zn * __fsqrt_rn(nu / chi2);           // ~StudentT(nu)
      float factor = (e >= 50 && e < 65) ? 1.5f : 1.0f;  // 1 + 5*sigma
      out[e] = tstd * __expf(s_t) * factor;
    }
  }
  return chunkTot;
}

// ---------- Kernel C: rescan tiles (TDM double-buffered) + emit ----------
__global__ __launch_bounds__(kThreads)
void sv_scan_emit(const float* __restrict__ eps_raw,
                  const float* __restrict__ log_theta,
                  const float* __restrict__ offs,
                  float* __restrict__ out, int n) {
  __shared__ float buf[2][kChunk];   // double-buffered raw eps staging
  __shared__ float sbuf[kChunk];
  __shared__ float wsum[kWaves];

  const int t    = threadIdx.x;
  const int lane = t & 31;
  const int lj   = lane & 15;
  const int hi   = lane >> 4;
  const int tileBase = blockIdx.x * kTile;

  const float nu     = __expf(log_theta[0]) + 1.0f;
  const float invTau = 1.0f / (__expf(log_theta[1]) + 1.0f);
  const float whC    = 2.0f / (9.0f * nu);
  const float whS    = __fsqrt_rn(whC);

  // Triangular A-matrix (16x4 K-slices) with the 1/tau scale folded in.
  v2f A[4];
#pragma unroll
  for (int c = 0; c < 4; ++c) {
    const int k0 = 4 * c + 2 * hi;
    A[c].x = (k0     <= lj) ? invTau : 0.0f;   // (invTau*L)[lj][k0]
    A[c].y = (k0 + 1 <= lj) ? invTau : 0.0f;   // (invTau*L)[lj][k0+1]
  }

  float blockOffset = offs[blockIdx.x];

#if SV_USE_TDM
  if (tileBase + kTile <= n) {   // full tile: checkless, TDM double-buffered
    const uint32_t ldsBase = sv_lds_offset(&buf[0][0]);
    if (t < 32) sv_tdm_load_chunk(ldsBase, eps_raw + tileBase);   // prime T(0)
    for (int it = 0; it < kIters; ++it) {
      const int sel = it & 1;
      if (t < 32) {              // wave 0 drives the TDM pipeline
        if (it + 1 < kIters) {
          sv_tdm_load_chunk(ldsBase + (uint32_t)((sel ^ 1) * kChunk * 4),
                            eps_raw + tileBase + (it + 1) * kChunk);
          __builtin_amdgcn_s_wait_tensorcnt(1);   // T(it) done, T(it+1) in flight
        } else {
          __builtin_amdgcn_s_wait_tensorcnt(0);   // drain last transfer
        }
      }
      __syncthreads();           // publish TDM-written LDS to all waves
      blockOffset += sv_chunk_compute<true>(buf[sel], sbuf, wsum, A, blockOffset,
                                            tileBase + it * kChunk, n,
                                            nu, whC, whS, out);
      __syncthreads();           // buffers reusable next iteration
    }
    return;
  }
#endif
  // partial / fallback tile: staged via VGPRs, zero-padded (scan-neutral).
  for (int it = 0; it < kIters; ++it) {
    const int chunkBase = tileBase + it * kChunk;
#pragma unroll
    for (int q = 0; q < kChunk / kThreads; ++q) {
      int e = chunkBase + q * kThreads + t;
      buf[0][q * kThreads + t] = (e < n) ? eps_raw[e] : 0.0f;   // raw eps
    }
    __syncthreads();
    blockOffset += sv_chunk_compute<false>(buf[0], sbuf, wsum, A, blockOffset,
                                           chunkBase, n, nu, whC, whS, out);
    __syncthreads();
  }
}

extern "C" void kernel_launch(void* const* d_in, const int* in_sizes, int n_in,
                              void* d_out, int out_size, void* d_ws, size_t ws_size,
                              hipStream_t stream) {
  const float* log_theta = (const float*)d_in[0];
  const float* eps_raw   = (const float*)d_in[1];
  float* out = (float*)d_out;
  const int n  = in_sizes[1];                 // T + 1
  const int nb = (n + kTile - 1) / kTile;     // 513 tiles for T=16,777,216

  float* part = (float*)d_ws;
  float* offs = part + ((nb + 255) & ~255);

  sv_tile_reduce<<<nb, kThreads, 0, stream>>>(eps_raw, log_theta, part, n);
  sv_scan_partials<<<1, 1024, 0, stream>>>(part, offs, nb);
  sv_scan_emit<<<nb, kThreads, 0, stream>>>(eps_raw, log_theta, offs, out, n);
}